// PatchTSTRopeAttention_73461120631351
// MI455X (gfx1250) — compile-verified
//
#include <hip/hip_runtime.h>
#include <hip/hip_bf16.h>

typedef float v2f __attribute__((ext_vector_type(2)));
typedef float v8f __attribute__((ext_vector_type(8)));

#define S_LEN   2048
#define EMB     512
#define NHEADS  8
#define HDIM    64
#define BATCH   4
#define BH      (BATCH * NHEADS)      // 32
#define ROWS    (BATCH * S_LEN)       // 8192
#define NEG_BIG (-3.0e38f)

// ---------------------------------------------------------------------------
// Kernel 1: fused QKV projection. One wave -> 64x16 tile of Q,K,V (register
// blocked: 4 row sub-tiles x 3 matrices = 12 WMMAs per K=4 step, sharing the
// three B fragments across 4 A fragments). FP32 WMMA 16x16x4.
// Q gets bias + 1/sqrt(D) scaling; outputs stored head-major [BH][S][D].
// ---------------------------------------------------------------------------
__global__ __launch_bounds__(128) void qkv_gemm_kernel(
    const float* __restrict__ X,
    const float* __restrict__ Wq, const float* __restrict__ bq,
    const float* __restrict__ Wk, const float* __restrict__ bk,
    const float* __restrict__ Wv, const float* __restrict__ bv,
    float* __restrict__ Qb, float* __restrict__ Kb, float* __restrict__ Vb)
{
    const int lane = threadIdx.x & 31;
    const int wg   = blockIdx.x * 4 + (threadIdx.x >> 5);   // 0..4095
    const int row0 = (wg >> 5) * 64;          // 128 row tiles of 64
    const int col0 = (wg & 31) * 16;          // 32 col tiles of 16
    const int m    = lane & 15;
    const int koff = (lane >> 4) << 1;        // 0 or 2
    const int hi8  = (lane >> 4) << 3;        // 0 or 8

    v8f cq[4] = {}, ck[4] = {}, cv[4] = {};

    for (int k = 0; k < EMB; k += 4) {
        v2f a[4];
#pragma unroll
        for (int r = 0; r < 4; ++r)
            a[r] = *(const v2f*)(X + (size_t)(row0 + r * 16 + m) * EMB + k + koff);
        const int w0 = (k + koff) * EMB + col0 + m;
        v2f fq, fk, fv;
        fq.x = Wq[w0]; fq.y = Wq[w0 + EMB];
        fk.x = Wk[w0]; fk.y = Wk[w0 + EMB];
        fv.x = Wv[w0]; fv.y = Wv[w0 + EMB];
#pragma unroll
        for (int r = 0; r < 4; ++r) {
            cq[r] = __builtin_amdgcn_wmma_f32_16x16x4_f32(false, a[r], false, fq, (short)0, cq[r], false, false);
            ck[r] = __builtin_amdgcn_wmma_f32_16x16x4_f32(false, a[r], false, fk, (short)0, ck[r], false, false);
            cv[r] = __builtin_amdgcn_wmma_f32_16x16x4_f32(false, a[r], false, fv, (short)0, cv[r], false, false);
        }
    }

    const int col = col0 + m;
    const int h = col >> 6, d = col & 63;
    const float biasq = bq[col], biask = bk[col], biasv = bv[col];
#pragma unroll
    for (int r = 0; r < 4; ++r)
#pragma unroll
        for (int i = 0; i < 8; ++i) {
            const int row = row0 + r * 16 + i + hi8;
            const int b = row >> 11, s = row & (S_LEN - 1);
            const size_t o = ((size_t)((b * NHEADS + h) * S_LEN + s)) * HDIM + d;
            Qb[o] = (cq[r][i] + biasq) * 0.125f;   // HEAD_DIM^-0.5
            Kb[o] = ck[r][i] + biask;
            Vb[o] = cv[r][i] + biasv;
        }
}

// ---------------------------------------------------------------------------
// Kernel 2: partial RoPE (rope_percent = 0.5 -> 16 rotated pairs of 32).
// In-place on head-major Q and K.
// ---------------------------------------------------------------------------
__global__ __launch_bounds__(256) void rope_kernel(float* __restrict__ Qb,
                                                   float* __restrict__ Kb)
{
    const int idx = blockIdx.x * blockDim.x + threadIdx.x;  // BH*S*32 threads
    const int d  = idx & 31;
    const int s  = (idx >> 5) & (S_LEN - 1);
    const int bh = idx >> 16;
    if (bh >= BH) return;
    if (d >= 16) return;                                    // 'nope' channels: identity

    const float fraction = (2.0f * (float)d) / 64.0f;
    const float inv_ts = __expf(-fraction * 9.2103403719761836f); // ln(10000)
    const float ang = (float)s * inv_ts;
    const float sn = __sinf(ang), cs = __cosf(ang);

    const size_t base = ((size_t)bh * S_LEN + s) * HDIM;
    float q1 = Qb[base + d], q2 = Qb[base + d + 32];
    Qb[base + d]      = q1 * cs - q2 * sn;
    Qb[base + d + 32] = q2 * cs + q1 * sn;
    float k1 = Kb[base + d], k2 = Kb[base + d + 32];
    Kb[base + d]      = k1 * cs - k2 * sn;
    Kb[base + d + 32] = k2 * cs + k1 * sn;
}

// ---------------------------------------------------------------------------
// Kernel 3: causal flash attention, fp32 WMMA. One wave per (head, 32-row
// query block = 2 sub-tiles): K and V fragments loaded once per k-tile are
// reused by both sub-tiles, doubling wmma per vmem. Online softmax via
// wave32 shfl_xor over 16-lane groups; P transposed through LDS.
// ---------------------------------------------------------------------------
__global__ __launch_bounds__(128) void attn_kernel(
    const float* __restrict__ Qb, const float* __restrict__ Kb,
    const float* __restrict__ Vb, float* __restrict__ Ab)
{
    __shared__ float ldsP[4][32 * 16];

    const int lane = threadIdx.x & 31;
    const int w    = threadIdx.x >> 5;
    const int wg   = blockIdx.x * 4 + w;      // 0..2047
    const int bh   = wg >> 6;                 // 32 heads
    const int q0   = (wg & 63) * 32;          // 64 query blocks of 32 rows
    const int m    = lane & 15;
    const int koff = (lane >> 4) << 1;
    const int hi8  = (lane >> 4) << 3;

    const float* Qh = Qb + (size_t)bh * S_LEN * HDIM;
    const float* Kh = Kb + (size_t)bh * S_LEN * HDIM;
    const float* Vh = Vb + (size_t)bh * S_LEN * HDIM;

    // Preload Q for both sub-tiles as A-fragments (D=64 -> 16 steps of K=4).
    v2f qf[2][16];
#pragma unroll
    for (int qs = 0; qs < 2; ++qs)
#pragma unroll
        for (int f = 0; f < 16; ++f)
            qf[qs][f] = *(const v2f*)(Qh + (size_t)(q0 + qs * 16 + m) * HDIM + f * 4 + koff);

    v8f acc[2][4] = {};                       // O accumulators: 2 x 4 x (16x16)
    float mrow[2][8], lrow[2][8];
#pragma unroll
    for (int qs = 0; qs < 2; ++qs)
#pragma unroll
        for (int i = 0; i < 8; ++i) { mrow[qs][i] = NEG_BIG; lrow[qs][i] = 0.0f; }

    const int dtile  = q0 >> 4;               // first tile needing the mask
    const int ntiles = dtile + 2;             // causal: tiles up to q0+16
    for (int t = 0; t < ntiles; ++t) {
        const int k0 = t * 16;

        // S = Q @ K^T : 16 K-fragments, each feeding both sub-tiles (32 WMMAs)
        v8f sc[2];
        sc[0] = {}; sc[1] = {};
#pragma unroll
        for (int f = 0; f < 16; ++f) {
            v2f kf = *(const v2f*)(Kh + (size_t)(k0 + m) * HDIM + f * 4 + koff);
            sc[0] = __builtin_amdgcn_wmma_f32_16x16x4_f32(false, qf[0][f], false, kf, (short)0, sc[0], false, false);
            sc[1] = __builtin_amdgcn_wmma_f32_16x16x4_f32(false, qf[1][f], false, kf, (short)0, sc[1], false, false);
        }

        const bool masked = (t >= dtile);     // wave-uniform
#pragma unroll
        for (int qs = 0; qs < 2; ++qs) {
            if (masked) {                     // per-lane causal select
                const int colk = k0 + m;
#pragma unroll
                for (int i = 0; i < 8; ++i)
                    if (colk > q0 + qs * 16 + i + hi8) sc[qs][i] = NEG_BIG;
            }
            // Online softmax: rows striped N-across-16-lanes in C layout.
#pragma unroll
            for (int i = 0; i < 8; ++i) {
                float rm = sc[qs][i];
                rm = fmaxf(rm, __shfl_xor(rm, 1, 32));
                rm = fmaxf(rm, __shfl_xor(rm, 2, 32));
                rm = fmaxf(rm, __shfl_xor(rm, 4, 32));
                rm = fmaxf(rm, __shfl_xor(rm, 8, 32));
                const float mnew  = fmaxf(mrow[qs][i], rm);
                const float alpha = __expf(mrow[qs][i] - mnew);
                const float pv    = __expf(sc[qs][i] - mnew);
                float rs = pv;
                rs += __shfl_xor(rs, 1, 32);
                rs += __shfl_xor(rs, 2, 32);
                rs += __shfl_xor(rs, 4, 32);
                rs += __shfl_xor(rs, 8, 32);
                lrow[qs][i] = lrow[qs][i] * alpha + rs;
                mrow[qs][i] = mnew;
#pragma unroll
                for (int j = 0; j < 4; ++j) acc[qs][j][i] *= alpha;
                // stash unnormalized p back into sc for the LDS transpose
                sc[qs][i] = pv;
            }
            // Transpose P (C layout) into row-major LDS.
#pragma unroll
            for (int i = 0; i < 8; ++i)
                ldsP[w][(qs * 16 + i + hi8) * 16 + m] = sc[qs][i];
        }
        asm volatile("s_wait_dscnt 0x0" ::: "memory");  // same-wave LDS RAW

        // O += P @ V : V fragments shared by both sub-tiles (32 WMMAs)
#pragma unroll
        for (int kk = 0; kk < 16; kk += 4) {
            v2f pf0 = *(const v2f*)(&ldsP[w][(m)      * 16 + kk + koff]);
            v2f pf1 = *(const v2f*)(&ldsP[w][(16 + m) * 16 + kk + koff]);
            const float* vrow = Vh + (size_t)(k0 + kk + koff) * HDIM + m;
#pragma unroll
            for (int j = 0; j < 4; ++j) {
                v2f vf;
                vf.x = vrow[j * 16];
                vf.y = vrow[j * 16 + HDIM];
                acc[0][j] = __builtin_amdgcn_wmma_f32_16x16x4_f32(false, pf0, false, vf, (short)0, acc[0][j], false, false);
                acc[1][j] = __builtin_amdgcn_wmma_f32_16x16x4_f32(false, pf1, false, vf, (short)0, acc[1][j], false, false);
            }
        }
    }

    // Epilogue: divide by softmax denominator, write back to [B][S][E].
    const int b = bh >> 3, h = bh & 7;
#pragma unroll
    for (int qs = 0; qs < 2; ++qs) {
        float inv[8];
#pragma unroll
        for (int i = 0; i < 8; ++i) inv[i] = 1.0f / lrow[qs][i];
#pragma unroll
        for (int j = 0; j < 4; ++j)
#pragma unroll
            for (int i = 0; i < 8; ++i) {
                const int row = q0 + qs * 16 + i + hi8;
                Ab[((size_t)(b * S_LEN + row)) * EMB + h * HDIM + j * 16 + m] =
                    acc[qs][j][i] * inv[i];
            }
    }
}

// ---------------------------------------------------------------------------
// Kernel 4: output projection out = attn @ Wo + bo. 64x32 tile per wave:
// 8 WMMAs per K=4 step over 4 A-frags x 2 B-frags.
// ---------------------------------------------------------------------------
__global__ __launch_bounds__(128) void out_gemm_kernel(
    const float* __restrict__ Ain, const float* __restrict__ Wo,
    const float* __restrict__ bo, float* __restrict__ Out)
{
    const int lane = threadIdx.x & 31;
    const int wg   = blockIdx.x * 4 + (threadIdx.x >> 5);   // 0..2047
    const int row0 = (wg >> 4) * 64;          // 128 row tiles of 64
    const int col0 = (wg & 15) * 32;          // 16 col tiles of 32
    const int m    = lane & 15;
    const int koff = (lane >> 4) << 1;
    const int hi8  = (lane >> 4) << 3;

    v8f c[4][2] = {};
    for (int k = 0; k < EMB; k += 4) {
        v2f a[4];
#pragma unroll
        for (int r = 0; r < 4; ++r)
            a[r] = *(const v2f*)(Ain + (size_t)(row0 + r * 16 + m) * EMB + k + koff);
        v2f bf[2];
#pragma unroll
        for (int n = 0; n < 2; ++n) {
            const int w0 = (k + koff) * EMB + col0 + n * 16 + m;
            bf[n].x = Wo[w0]; bf[n].y = Wo[w0 + EMB];
        }
#pragma unroll
        for (int r = 0; r < 4; ++r)
#pragma unroll
            for (int n = 0; n < 2; ++n)
                c[r][n] = __builtin_amdgcn_wmma_f32_16x16x4_f32(false, a[r], false, bf[n], (short)0, c[r][n], false, false);
    }
#pragma unroll
    for (int n = 0; n < 2; ++n) {
        const float bias = bo[col0 + n * 16 + m];
#pragma unroll
        for (int r = 0; r < 4; ++r)
#pragma unroll
            for (int i = 0; i < 8; ++i)
                Out[(size_t)(row0 + r * 16 + i + hi8) * EMB + col0 + n * 16 + m] = c[r][n][i] + bias;
    }
}

// ---------------------------------------------------------------------------
extern "C" void kernel_launch(void* const* d_in, const int* in_sizes, int n_in,
                              void* d_out, int out_size, void* d_ws, size_t ws_size,
                              hipStream_t stream) {
    const float* X  = (const float*)d_in[0];
    const float* Wq = (const float*)d_in[1];
    const float* bq = (const float*)d_in[2];
    const float* Wk = (const float*)d_in[3];
    const float* bk = (const float*)d_in[4];
    const float* Wv = (const float*)d_in[5];
    const float* bv = (const float*)d_in[6];
    const float* Wo = (const float*)d_in[7];
    const float* bo = (const float*)d_in[8];
    float* out = (float*)d_out;

    const size_t BUF = (size_t)BH * S_LEN * HDIM;   // 4,194,304 floats
    float* Qb = (float*)d_ws;
    float* Kb = Qb + BUF;
    float* Vb = Kb + BUF;
    float* Ab = Vb + BUF;

    // 128x32 tiles (64x16 each), 4 waves/block
    qkv_gemm_kernel<<<1024, 128, 0, stream>>>(X, Wq, bq, Wk, bk, Wv, bv, Qb, Kb, Vb);
    // BH*S*32 threads
    rope_kernel<<<(BH * S_LEN * 32) / 256, 256, 0, stream>>>(Qb, Kb);
    // 32 heads x 64 query blocks (32 rows), 4 waves/block
    attn_kernel<<<512, 128, 0, stream>>>(Qb, Kb, Vb, Ab);
    // 128x16 tiles (64x32 each), 4 waves/block
    out_gemm_kernel<<<512, 128, 0, stream>>>(Ab, Wo, bo, out);
}